// EncoderMod_74380243632346
// MI455X (gfx1250) — compile-verified
//
#include <hip/hip_runtime.h>

typedef float v2f __attribute__((ext_vector_type(2)));
typedef float v8f __attribute__((ext_vector_type(8)));

#define N_NODES 50000
#define N_EDGES 800000

__global__ void zero_f32(float* __restrict__ p, int n) {
  int i = blockIdx.x * blockDim.x + threadIdx.x;
  if (i < n) p[i] = 0.0f;
}

__global__ void degree_kernel(const int* __restrict__ dst, float* __restrict__ deg, int E) {
  int e = blockIdx.x * blockDim.x + threadIdx.x;
  if (e < E) atomicAdd(&deg[dst[e]], 1.0f);
}

__global__ void dinv_kernel(float* __restrict__ deg, float* __restrict__ dinv, int n) {
  int i = blockIdx.x * blockDim.x + threadIdx.x;
  if (i < n) {
    float d = deg[i] + 1.0f;   // +1 for self loop
    deg[i] = d;
    dinv[i] = 1.0f / sqrtf(d); // deg ** -0.5
  }
}

// H[M,F] = X[M,K] @ W[K,F] in exact fp32 via V_WMMA_F32_16X16X4_F32.
// One wave per 16x16 output tile. blockDim.x = 32*(F/16), grid.x = M/16.
// A fragment layout (16x4 f32): lanes 0-15 -> K=0,1 ; lanes 16-31 -> K=2,3.
// D layout (16x16 f32): VGPR r: lanes 0-15 -> M=r, lanes 16-31 -> M=r+8.
__global__ void gemm_wmma_f32(const float* __restrict__ X, const float* __restrict__ W,
                              float* __restrict__ H, int K, int F) {
  __shared__ float As[16 * 132];  // 16 rows, stride 132 dwords: 16B aligned + bank-conflict free
  const int lane  = threadIdx.x & 31;
  const int ntile = threadIdx.x >> 5;   // wave id == N tile index
  const int mtile = blockIdx.x;
  const size_t xbase = (size_t)mtile * 16 * K;

  // Cooperatively stage the 16xK (K=128) A tile into LDS once per block.
  for (int idx = threadIdx.x * 4; idx < 16 * 128; idx += blockDim.x * 4) {
    const int r = idx >> 7, c = idx & 127;
    const float4 v = *(const float4*)(X + xbase + (size_t)r * K + c);
    *(float4*)(&As[r * 132 + c]) = v;
  }
  __syncthreads();

  const int half = lane >> 4;      // 0 or 1
  const int m    = lane & 15;
  const int koff = half * 2;       // 0 or 2
  const int col  = ntile * 16 + m;
  const float* as = &As[m * 132];

  v8f acc = {};
#pragma unroll 4
  for (int k = 0; k < K; k += 4) {
    v2f a = *(const v2f*)(as + k + koff);          // ds_load_b64, conflict-free
    v2f b;
    b.x = W[(size_t)(k + koff) * F + col];
    b.y = W[(size_t)(k + koff + 1) * F + col];
    acc = __builtin_amdgcn_wmma_f32_16x16x4_f32(
        /*neg_a=*/false, a, /*neg_b=*/false, b,
        /*c_mod=*/(short)0, acc, /*reuse_a=*/false, /*reuse_b=*/false);
  }

  const int orow0 = mtile * 16 + half * 8;
#pragma unroll
  for (int r = 0; r < 8; ++r)
    H[(size_t)(orow0 + r) * F + col] = acc[r];
}

// One wave per edge: coalesced gather of h[src] row, scaled atomic scatter to agg[dst].
__global__ void scatter_kernel(const int* __restrict__ src, const int* __restrict__ dst,
                               const float* __restrict__ h, const float* __restrict__ dinv,
                               float* __restrict__ agg, int E, int F) {
  const int g = blockIdx.x * blockDim.x + threadIdx.x;
  const int e = g >> 5;
  const int lane = g & 31;
  if (e >= E) return;
  const int s = src[e], d = dst[e];
  const float norm = dinv[s] * dinv[d];
  const float* hs = h + (size_t)s * F;
  float* ad = agg + (size_t)d * F;
  for (int c = lane; c < F; c += 32)
    atomicAdd(&ad[c], hs[c] * norm);
}

// out = relu(agg + h * (1/deg) + b)   (out may alias agg; elementwise-safe)
__global__ void finish_kernel(const float* __restrict__ agg, const float* __restrict__ h,
                              const float* __restrict__ deg, const float* __restrict__ b,
                              float* __restrict__ out, int total, int fmask, int fshift) {
  const int i = blockIdx.x * blockDim.x + threadIdx.x;
  if (i >= total) return;
  const int node = i >> fshift;
  const int c = i & fmask;
  const float v = agg[i] + h[i] * (1.0f / deg[node]) + b[c];
  out[i] = v > 0.0f ? v : 0.0f;
}

extern "C" void kernel_launch(void* const* d_in, const int* in_sizes, int n_in,
                              void* d_out, int out_size, void* d_ws, size_t ws_size,
                              hipStream_t stream) {
  const float* x   = (const float*)d_in[0];
  const int*   ei  = (const int*)d_in[1];
  const float* W1  = (const float*)d_in[2];
  const float* b1  = (const float*)d_in[3];
  const float* W2  = (const float*)d_in[4];
  const float* b2  = (const float*)d_in[5];
  float* out = (float*)d_out;

  const int N = N_NODES, E = N_EDGES;
  const int* src  = ei;       // edge_index[0]
  const int* dstv = ei + E;   // edge_index[1]

  // Workspace layout (floats):
  float* ws   = (float*)d_ws;
  float* deg  = ws;                       // N
  float* dinv = ws + N;                   // N
  float* bufA = ws + 2 * (size_t)N;       // N*128 : h1, later reused as h2 | agg2
  float* bufB = bufA + (size_t)N * 128;   // N*128 : agg1, then relu'd h1 in place
  float* h2   = bufA;                     // N*64
  float* agg2 = bufA + (size_t)N * 64;    // N*64

  const int T = 256;
  zero_f32<<<(N + T - 1) / T, T, 0, stream>>>(deg, N);
  zero_f32<<<(N * 128 + T - 1) / T, T, 0, stream>>>(bufB, N * 128);
  degree_kernel<<<(E + T - 1) / T, T, 0, stream>>>(dstv, deg, E);
  dinv_kernel<<<(N + T - 1) / T, T, 0, stream>>>(deg, dinv, N);

  // ---- Layer 1: F = 128 ----
  gemm_wmma_f32<<<N / 16, 32 * (128 / 16), 0, stream>>>(x, W1, bufA, 128, 128);
  scatter_kernel<<<(E * 32 + T - 1) / T, T, 0, stream>>>(src, dstv, bufA, dinv, bufB, E, 128);
  finish_kernel<<<(N * 128 + T - 1) / T, T, 0, stream>>>(bufB, bufA, deg, b1, bufB,
                                                         N * 128, 127, 7);

  // ---- Layer 2: F = 64 ----
  gemm_wmma_f32<<<N / 16, 32 * (64 / 16), 0, stream>>>(bufB, W2, h2, 128, 64);
  zero_f32<<<(N * 64 + T - 1) / T, T, 0, stream>>>(agg2, N * 64);
  scatter_kernel<<<(E * 32 + T - 1) / T, T, 0, stream>>>(src, dstv, h2, dinv, agg2, E, 64);
  finish_kernel<<<(N * 64 + T - 1) / T, T, 0, stream>>>(agg2, h2, deg, b2, out,
                                                        N * 64, 63, 6);
}